// SVD_series_decomp_5497558138982
// MI455X (gfx1250) — compile-verified
//
#include <hip/hip_runtime.h>
#include <hip/hip_bf16.h>

typedef __attribute__((ext_vector_type(2))) float v2f;
typedef __attribute__((ext_vector_type(8))) float v8f;

#define NROWS  131072   // 64 * 2048
#define CDIM   64
#define TOPK   5

// ---------------------------------------------------------------------------
// Kernel 0: zero the Gram accumulator
// ---------------------------------------------------------------------------
__global__ __launch_bounds__(256) void zero_gram_kernel(float* __restrict__ G) {
    int i = blockIdx.x * 256 + threadIdx.x;
    if (i < CDIM * CDIM) G[i] = 0.0f;
}

// ---------------------------------------------------------------------------
// Kernel 1: G = M^T M  (64x64), WMMA f32 16x16x4, partials via atomicAdd
// Each block stages 64 rows x 64 cols into LDS; 8 waves each own two 16x16
// tiles of the 4x4 tile grid of G.
// ---------------------------------------------------------------------------
__global__ __launch_bounds__(256) void gram_kernel(const float* __restrict__ M,
                                                   float* __restrict__ G,
                                                   int rows_per_block) {
    __shared__ float lds[64 * 64];   // 16 KB tile of M
    const int lane = threadIdx.x & 31;
    const int wave = threadIdx.x >> 5;      // 0..7
    const int m    = lane & 15;             // row within A tile / col within B tile
    const int kh   = (lane >> 4) * 2;       // K half select (0 or 2)
    const int row0 = blockIdx.x * rows_per_block;

    v8f acc0 = {};
    v8f acc1 = {};

    for (int t = 0; t < rows_per_block; t += 64) {
        // cooperative float4 load: 64x64 floats = 1024 float4
        const float4* src = (const float4*)(M + (size_t)(row0 + t) * CDIM);
        float4* dst = (float4*)lds;
        #pragma unroll
        for (int i = 0; i < 4; ++i)
            dst[threadIdx.x + i * 256] = src[threadIdx.x + i * 256];
        __syncthreads();

        #pragma unroll 1
        for (int p = 0; p < 2; ++p) {
            const int tile = wave * 2 + p;
            const int ti = tile >> 2;       // A-tile (rows of G)
            const int tj = tile & 3;        // B-tile (cols of G)
            v8f acc = (p == 0) ? acc0 : acc1;
            #pragma unroll
            for (int kc = 0; kc < 16; ++kc) {
                // A(m,k) = M[row k][col ti*16+m]  (transposed gather from LDS)
                v2f a, b;
                a.x = lds[(kc * 4 + kh + 0) * 64 + ti * 16 + m];
                a.y = lds[(kc * 4 + kh + 1) * 64 + ti * 16 + m];
                // B(k,n) = M[row k][col tj*16+n]
                b.x = lds[(kc * 4 + kh + 0) * 64 + tj * 16 + m];
                b.y = lds[(kc * 4 + kh + 1) * 64 + tj * 16 + m];
                acc = __builtin_amdgcn_wmma_f32_16x16x4_f32(
                        false, a, false, b, (short)0, acc, false, false);
            }
            if (p == 0) acc0 = acc; else acc1 = acc;
        }
        __syncthreads();
    }

    // accumulate partial G tiles into global
    const int mh = (lane >> 4) * 8;
    #pragma unroll 1
    for (int p = 0; p < 2; ++p) {
        const int tile = wave * 2 + p;
        const int ti = tile >> 2;
        const int tj = tile & 3;
        v8f acc = (p == 0) ? acc0 : acc1;
        #pragma unroll
        for (int r = 0; r < 8; ++r) {
            const int gi = ti * 16 + r + mh;
            const int gj = tj * 16 + (lane & 15);
            atomicAdd(&G[gi * 64 + gj], acc[r]);
        }
    }
}

// ---------------------------------------------------------------------------
// Kernel 2: top-5 eigenvectors of G via block subspace iteration + MGS,
// then P = Q Q^T (projector onto top-5 right singular subspace).
// Single workgroup of 64 threads; trivially cheap.
// ---------------------------------------------------------------------------
__global__ __launch_bounds__(64) void topk_projector_kernel(const float* __restrict__ G,
                                                            float* __restrict__ P) {
    __shared__ float g[64 * 64];
    __shared__ float q[64 * TOPK];
    __shared__ float z[64 * TOPK];
    __shared__ float red[64];
    const int i = threadIdx.x;

    for (int j = 0; j < 64; ++j) g[i * 64 + j] = G[i * 64 + j];
    // deterministic pseudo-random init
    #pragma unroll
    for (int c = 0; c < TOPK; ++c)
        q[i * TOPK + c] = sinf(0.37f * (float)i + 1.71f * (float)c + 0.5f);
    __syncthreads();

    for (int it = 0; it < 48; ++it) {
        // z = G * q  (row i per thread)
        #pragma unroll
        for (int c = 0; c < TOPK; ++c) {
            float s = 0.0f;
            for (int j = 0; j < 64; ++j) s += g[i * 64 + j] * q[j * TOPK + c];
            z[i * TOPK + c] = s;
        }
        __syncthreads();
        // modified Gram-Schmidt: orthonormalize columns of z into q
        for (int c = 0; c < TOPK; ++c) {
            for (int p = 0; p < c; ++p) {
                red[i] = z[i * TOPK + c] * q[i * TOPK + p];
                __syncthreads();
                for (int s = 32; s > 0; s >>= 1) {
                    if (i < s) red[i] += red[i + s];
                    __syncthreads();
                }
                const float dot = red[0];
                __syncthreads();
                z[i * TOPK + c] -= dot * q[i * TOPK + p];
                __syncthreads();
            }
            red[i] = z[i * TOPK + c] * z[i * TOPK + c];
            __syncthreads();
            for (int s = 32; s > 0; s >>= 1) {
                if (i < s) red[i] += red[i + s];
                __syncthreads();
            }
            const float nrm = sqrtf(red[0]) + 1e-20f;
            __syncthreads();
            q[i * TOPK + c] = z[i * TOPK + c] / nrm;
            __syncthreads();
        }
    }

    // P = Q Q^T
    for (int j = 0; j < 64; ++j) {
        float s = 0.0f;
        #pragma unroll
        for (int c = 0; c < TOPK; ++c) s += q[i * TOPK + c] * q[j * TOPK + c];
        P[i * 64 + j] = s;
    }
}

// ---------------------------------------------------------------------------
// Kernel 3: trend = M * P, season = x - trend. WMMA f32 16x16x4.
// Block = 256 threads = 8 waves; each wave owns a 16-row x 64-col slab.
// P (16 KB) + 128-row M tile (32 KB) staged in LDS.
// ---------------------------------------------------------------------------
__global__ __launch_bounds__(256) void reconstruct_kernel(const float* __restrict__ M,
                                                          const float* __restrict__ P,
                                                          float* __restrict__ trend,
                                                          float* __restrict__ season) {
    __shared__ float plds[64 * 64];     // 16 KB
    __shared__ float alds[128 * 64];    // 32 KB
    const int lane = threadIdx.x & 31;
    const int wave = threadIdx.x >> 5;          // 0..7
    const int m    = lane & 15;
    const int kh   = (lane >> 4) * 2;
    const size_t row0 = (size_t)blockIdx.x * 128;   // 128 rows per block

    // stage P
    {
        const float4* src = (const float4*)P;
        float4* dst = (float4*)plds;
        #pragma unroll
        for (int i = 0; i < 4; ++i)
            dst[threadIdx.x + i * 256] = src[threadIdx.x + i * 256];
    }
    // stage 128x64 tile of M (2048 float4)
    {
        const float4* src = (const float4*)(M + row0 * CDIM);
        float4* dst = (float4*)alds;
        #pragma unroll
        for (int i = 0; i < 8; ++i)
            dst[threadIdx.x + i * 256] = src[threadIdx.x + i * 256];
    }
    __syncthreads();

    v8f acc[4] = {{}, {}, {}, {}};
    const int abase = wave * 16 * 64;           // this wave's 16 rows in LDS

    #pragma unroll
    for (int kc = 0; kc < 16; ++kc) {
        v2f a;
        a.x = alds[abase + m * 64 + kc * 4 + kh + 0];
        a.y = alds[abase + m * 64 + kc * 4 + kh + 1];
        #pragma unroll
        for (int tj = 0; tj < 4; ++tj) {
            v2f b;
            b.x = plds[(kc * 4 + kh + 0) * 64 + tj * 16 + m];
            b.y = plds[(kc * 4 + kh + 1) * 64 + tj * 16 + m];
            acc[tj] = __builtin_amdgcn_wmma_f32_16x16x4_f32(
                        false, a, false, b, (short)0, acc[tj], false, false);
        }
    }

    // write trend + season (coalesced 64B segments per accumulator register)
    const int mh = (lane >> 4) * 8;
    #pragma unroll
    for (int tj = 0; tj < 4; ++tj) {
        #pragma unroll
        for (int r = 0; r < 8; ++r) {
            const int lm  = r + mh;                     // local row 0..15
            const size_t row = row0 + (size_t)(wave * 16 + lm);
            const int col = tj * 16 + (lane & 15);
            const float t = acc[tj][r];
            const float xv = alds[(wave * 16 + lm) * 64 + col];
            trend[row * CDIM + col]  = t;
            season[row * CDIM + col] = xv - t;
        }
    }
}

// ---------------------------------------------------------------------------
extern "C" void kernel_launch(void* const* d_in, const int* in_sizes, int n_in,
                              void* d_out, int out_size, void* d_ws, size_t ws_size,
                              hipStream_t stream) {
    const float* x = (const float*)d_in[0];
    float* trend  = (float*)d_out;
    float* season = (float*)d_out + (size_t)NROWS * CDIM;

    float* G = (float*)d_ws;                        // 64*64 floats
    float* P = (float*)d_ws + CDIM * CDIM;          // 64*64 floats

    // 1) zero Gram accumulator
    zero_gram_kernel<<<(CDIM * CDIM + 255) / 256, 256, 0, stream>>>(G);

    // 2) G = M^T M   (512 blocks x 256 rows each)
    const int gram_blocks = 512;
    const int rows_per_block = NROWS / gram_blocks;  // 256
    gram_kernel<<<gram_blocks, 256, 0, stream>>>(x, G, rows_per_block);

    // 3) top-5 projector P = Q Q^T
    topk_projector_kernel<<<1, 64, 0, stream>>>(G, P);

    // 4) trend = M P, season = x - trend  (128 rows per block)
    reconstruct_kernel<<<NROWS / 128, 256, 0, stream>>>(x, P, trend, season);
}